// Involution_88699664597251
// MI455X (gfx1250) — compile-verified
//
#include <hip/hip_runtime.h>

typedef float v2f __attribute__((ext_vector_type(2)));
typedef float v4f __attribute__((ext_vector_type(4)));
typedef float v8f __attribute__((ext_vector_type(8)));

#define CIN   256
#define CR    64
#define O2    784     // 49 * 16
#define HIMG  56
#define WIMG  56
#define HW    3136    // 56*56
#define NTILE_PER_B 196   // 3136 / 16

// ---------------------------------------------------------------------------
// Kernel 1: y[b][cr][hw] = ReLU(BN(w1 @ x))   (M=64, K=256, N=16 per wave)
// One wave per 16-pixel tile; 4 M-tiles accumulated, B operand shared.
// ---------------------------------------------------------------------------
__global__ __launch_bounds__(256) void invo_k1_pw1_bn_relu(
    const float* __restrict__ x,  const float* __restrict__ w1,
    const float* __restrict__ gm, const float* __restrict__ bt,
    const float* __restrict__ mu, const float* __restrict__ vr,
    float* __restrict__ y)
{
    const int lane = threadIdx.x & 31;
    const int wave = threadIdx.x >> 5;
    const int task = blockIdx.x * 8 + wave;      // exactly 784 tasks
    const int b    = task / NTILE_PER_B;
    const int n0   = (task % NTILE_PER_B) * 16;

    const int nl = lane & 15;                    // M-row / N-col within tile
    const int kh = lane >> 4;                    // K half (0/1)

    const float* xb = x + (size_t)b * CIN * HW;

    v8f acc[4];
    acc[0] = v8f{}; acc[1] = v8f{}; acc[2] = v8f{}; acc[3] = v8f{};

    #pragma unroll 4
    for (int k0 = 0; k0 < CIN; k0 += 4) {
        const int ka = k0 + 2 * kh;
        // B (K x N) : x rows ka, ka+1 at columns n0..n0+15
        v2f bm;
        bm.x = xb[(size_t)ka       * HW + n0 + nl];
        bm.y = xb[(size_t)(ka + 1) * HW + n0 + nl];
        // A (M x K) : w1 row (m0+nl), contiguous K pair -> 8B load
        v2f a0 = *(const v2f*)(w1 + (size_t)( 0 + nl) * CIN + ka);
        v2f a1 = *(const v2f*)(w1 + (size_t)(16 + nl) * CIN + ka);
        v2f a2 = *(const v2f*)(w1 + (size_t)(32 + nl) * CIN + ka);
        v2f a3 = *(const v2f*)(w1 + (size_t)(48 + nl) * CIN + ka);
        acc[0] = __builtin_amdgcn_wmma_f32_16x16x4_f32(false, a0, false, bm, (short)0, acc[0], false, false);
        acc[1] = __builtin_amdgcn_wmma_f32_16x16x4_f32(false, a1, false, bm, (short)0, acc[1], false, false);
        acc[2] = __builtin_amdgcn_wmma_f32_16x16x4_f32(false, a2, false, bm, (short)0, acc[2], false, false);
        acc[3] = __builtin_amdgcn_wmma_f32_16x16x4_f32(false, a3, false, bm, (short)0, acc[3], false, false);
    }

    // Epilogue: BN(eval) + ReLU, store y. C/D layout: VGPR r -> M = r + 8*kh, N = nl.
    const int col = n0 + nl;
    #pragma unroll
    for (int mt = 0; mt < 4; ++mt) {
        #pragma unroll
        for (int r = 0; r < 8; ++r) {
            const int row = mt * 16 + r + 8 * kh;
            const float s  = gm[row] * rsqrtf(vr[row] + 1e-5f);
            const float bb = bt[row] - mu[row] * s;
            float v = acc[mt][r] * s + bb;
            v = v > 0.f ? v : 0.f;
            y[((size_t)b * CR + row) * HW + col] = v;
        }
    }
}

// ---------------------------------------------------------------------------
// Kernel 2: fused conv2 (WMMA -> LDS) + vectorized involution.
// Block = 256 threads = 8 waves, one 16-pixel row-aligned tile.
// ---------------------------------------------------------------------------
__global__ __launch_bounds__(256) void invo_k2_pw2_involution(
    const float* __restrict__ x,  const float* __restrict__ y,
    const float* __restrict__ w2, const float* __restrict__ b2,
    float* __restrict__ out)
{
    __shared__ float y_lds[CR * 16];     // 4 KB : y tile [cr][p]
    __shared__ float w_lds[O2 * 16];     // 49 KB: per-pixel kernels [o2][p]

    const int tid  = threadIdx.x;
    const int lane = tid & 31;
    const int wave = tid >> 5;

    // Tile decode: 4 batches * 56 rows * 4 col-tiles = 896 blocks
    const int blk  = blockIdx.x;
    const int b    = blk / 224;
    const int rem  = blk % 224;
    const int h0   = rem / 4;
    const int ct   = rem % 4;
    const int col0 = (ct == 3) ? 40 : ct * 16;   // last tile overlaps; same values rewritten
    const int n0   = h0 * WIMG + col0;

    // Stage y tile into LDS (coalesced 16-wide rows)
    for (int q = tid; q < CR * 16; q += 256) {
        const int cr = q >> 4, j = q & 15;
        y_lds[q] = y[((size_t)b * CR + cr) * HW + n0 + j];
    }
    __syncthreads();

    // conv2: W[784][16] = w2 @ y_tile + b2, via f32 WMMA. 49 M-tiles over 8 waves.
    const int nl = lane & 15;
    const int kh = lane >> 4;
    for (int mt = wave; mt < 49; mt += 8) {
        const int m0 = mt * 16;
        v8f acc = v8f{};
        #pragma unroll
        for (int k0 = 0; k0 < CR; k0 += 4) {
            const int ka = k0 + 2 * kh;
            v2f a = *(const v2f*)(w2 + (size_t)(m0 + nl) * CR + ka);  // contiguous K pair
            v2f bm;
            bm.x = y_lds[ka       * 16 + nl];
            bm.y = y_lds[(ka + 1) * 16 + nl];
            acc = __builtin_amdgcn_wmma_f32_16x16x4_f32(false, a, false, bm, (short)0, acc, false, false);
        }
        #pragma unroll
        for (int r = 0; r < 8; ++r) {
            const int row = m0 + r + 8 * kh;
            w_lds[row * 16 + nl] = acc[r] + b2[row];
        }
    }
    __syncthreads();

    // ---------------- Involution, packed-math version ----------------------
    // Thread = (group g2, channel-quad cq, pixel-quad pq):
    //   computes out[g2*16+cq*4 .. +3][pixels 4pq..4pq+3] of this tile.
    // wA is a multiple of 4 -> halo span [wA-4, wA+7] is three aligned b128s.
    const int pq = tid & 3;
    const int cq = (tid >> 2) & 3;
    const int g2 = tid >> 4;
    const int wA = col0 + 4 * pq;
    const bool interior = (wA >= 4) && (wA <= WIMG - 8);

    const float* xg = x   + ((size_t)b * CIN + g2 * 16 + cq * 4) * HW;
    float*       og = out + ((size_t)b * CIN + g2 * 16 + cq * 4) * HW;

    v4f acc4[4];
    acc4[0] = v4f{}; acc4[1] = v4f{}; acc4[2] = v4f{}; acc4[3] = v4f{};

    for (int di = 0; di < 7; ++di) {
        const int hh = h0 + di - 3;
        if ((unsigned)hh >= (unsigned)HIMG) continue;   // uniform across block

        // per-pixel-quad kernel weights for this tap row: 7 aligned ds_b128
        v4f wvrow[7];
        #pragma unroll
        for (int dj = 0; dj < 7; ++dj)
            wvrow[dj] = *(const v4f*)&w_lds[(g2 * 49 + di * 7 + dj) * 16 + 4 * pq];

        #pragma unroll
        for (int c = 0; c < 4; ++c) {
            const float* rp = xg + (size_t)c * HW + hh * WIMG;
            float span[12];                       // x[wA-4 .. wA+7], kept in VGPRs
            if (interior) {
                *(v4f*)&span[0] = *(const v4f*)(rp + wA - 4);
                *(v4f*)&span[4] = *(const v4f*)(rp + wA);
                *(v4f*)&span[8] = *(const v4f*)(rp + wA + 4);
            } else {
                #pragma unroll
                for (int j = 0; j < 12; ++j) {
                    const int w = wA - 4 + j;
                    span[j] = ((unsigned)w < (unsigned)WIMG) ? rp[w] : 0.f;
                }
            }
            #pragma unroll
            for (int dj = 0; dj < 7; ++dj) {
                // pixel j (=0..3) at column wA+j, tap dj-3 -> span[j + dj + 1]
                v4f xv = {span[dj + 1], span[dj + 2], span[dj + 3], span[dj + 4]};
                acc4[c] += wvrow[dj] * xv;        // v_pk_fma_f32 pairs
            }
        }
    }

    #pragma unroll
    for (int c = 0; c < 4; ++c)
        *(v4f*)(og + (size_t)c * HW + n0 + 4 * pq) = acc4[c];   // aligned b128 store
}

// ---------------------------------------------------------------------------
extern "C" void kernel_launch(void* const* d_in, const int* in_sizes, int n_in,
                              void* d_out, int out_size, void* d_ws, size_t ws_size,
                              hipStream_t stream)
{
    const float* x  = (const float*)d_in[0];
    const float* w1 = (const float*)d_in[1];
    const float* gm = (const float*)d_in[2];
    const float* bt = (const float*)d_in[3];
    const float* mu = (const float*)d_in[4];
    const float* vr = (const float*)d_in[5];
    const float* w2 = (const float*)d_in[6];
    const float* b2 = (const float*)d_in[7];
    float* out = (float*)d_out;
    float* y   = (float*)d_ws;     // 4 * 64 * 3136 * 4 B = 3.2 MB scratch

    // 784 wave-tasks (16-pixel tiles), 8 waves/block -> 98 blocks, exact cover
    invo_k1_pw1_bn_relu<<<98, 256, 0, stream>>>(x, w1, gm, bt, mu, vr, y);
    // 4 batches * 56 rows * 4 col tiles = 896 blocks
    invo_k2_pw2_involution<<<896, 256, 0, stream>>>(x, y, w2, b2, out);
}